// EnhancedGIN_79044578116198
// MI455X (gfx1250) — compile-verified
//
#include <hip/hip_runtime.h>
#include <math.h>

typedef __attribute__((ext_vector_type(2))) float v2f;
typedef __attribute__((ext_vector_type(8))) float v8f;

#define NN   100000
#define EE   1600000
#define INDIM 64
#define HID  128
#define LAT  64
#define GG   1000

__device__ __forceinline__ float gelu_exact(float x) {
    return 0.5f * x * (1.0f + erff(x * 0.70710678118654752f));
}

// ---- input BatchNorm (fused with layer-0 agg init):
//      out = BN(x); agg = (1+eps0)*BN(x) ----
__global__ __launch_bounds__(256) void bn_in_kernel(
    const float* __restrict__ x,
    const float* __restrict__ g, const float* __restrict__ b,
    const float* __restrict__ m, const float* __restrict__ v,
    const float* __restrict__ epsP,
    float* __restrict__ out, float* __restrict__ agg)
{
    int idx = blockIdx.x * blockDim.x + threadIdx.x;
    if (idx >= NN * INDIM) return;
    int j = idx & (INDIM - 1);
    float y = (x[idx] - m[j]) * rsqrtf(v[j] + 1e-5f) * g[j] + b[j];
    out[idx] = y;
    agg[idx] = (1.0f + epsP[0]) * y;
}

// ---- edge scatter-add: agg[dst] += h[src]; one wave covers one edge's
//      feature vector in float4 chunks (coalesced 512B gather + FP atomics) ----
__global__ __launch_bounds__(256) void edge_agg_kernel(
    const float* __restrict__ h, const int* __restrict__ src,
    const int* __restrict__ dst, float* __restrict__ agg,
    int dim, int dim4, int total)
{
    int idx = blockIdx.x * blockDim.x + threadIdx.x;
    if (idx >= total) return;
    int e = idx / dim4;
    int c = idx - e * dim4;
    int s = src[e];
    int d = dst[e];
    const float4 val = *(const float4*)(h + (size_t)s * dim + 4 * c);
    float* p = agg + (size_t)d * dim + 4 * c;
    unsafeAtomicAdd(p + 0, val.x);
    unsafeAtomicAdd(p + 1, val.y);
    unsafeAtomicAdd(p + 2, val.z);
    unsafeAtomicAdd(p + 3, val.w);
}

// ---- WMMA f32 GEMM + bias + BN + GELU (+ optional fused (1+eps)*y agg init)
//      out[N][128] = GELU(BN(A[N][K] @ W[K][128] + bias))
//      One wave per 16x16 tile; 8 waves/block -> 16 rows x 128 cols per block.
//      N = 100000 = 6250*16, so no M-boundary handling. ----
template<int K>
__global__ __launch_bounds__(256) void gemm_bn_gelu_kernel(
    const float* __restrict__ A,
    const float* __restrict__ W,
    const float* __restrict__ bias,
    const float* __restrict__ bn_g, const float* __restrict__ bn_b,
    const float* __restrict__ bn_m, const float* __restrict__ bn_v,
    float* __restrict__ out,
    float* __restrict__ aggOut, const float* __restrict__ epsP,
    int epsIdx, int writeAgg)
{
    const int lane = threadIdx.x & 31;
    const int wave = threadIdx.x >> 5;        // 0..7 -> tile column
    const int tileM = blockIdx.x;
    const int half = lane >> 4;               // 0 or 1
    const int mr   = lane & 15;

    const int rowA = tileM * 16 + mr;         // A row this lane loads
    const int colB = wave * 16 + mr;          // B/N column this lane loads

    const float* __restrict__ Arow = A + (size_t)rowA * K;

    v8f acc = {};
    #pragma unroll
    for (int k = 0; k < K; k += 4) {
        const int ka = k + 2 * half;          // lanes 0-15: K={k,k+1}; 16-31: K={k+2,k+3}
        v2f a, b;
        a.x = Arow[ka];
        a.y = Arow[ka + 1];
        b.x = W[(size_t)ka * HID + colB];
        b.y = W[(size_t)(ka + 1) * HID + colB];
        acc = __builtin_amdgcn_wmma_f32_16x16x4_f32(
            false, a, false, b, (short)0, acc, false, false);
    }

    const float escale = writeAgg ? (1.0f + epsP[epsIdx]) : 0.0f;

    // Epilogue: acc[r] is element (m = r + 8*half, n = lane&15) of the tile.
    #pragma unroll
    for (int r = 0; r < 8; ++r) {
        const int row = tileM * 16 + r + 8 * half;
        const int col = wave * 16 + mr;
        float y = acc[r] + bias[col];
        y = (y - bn_m[col]) * rsqrtf(bn_v[col] + 1e-5f) * bn_g[col] + bn_b[col];
        y = gelu_exact(y);
        const size_t o = (size_t)row * HID + col;
        out[o] = y;
        if (writeAgg) aggOut[o] = escale * y;
    }
}

// ---- pooling ----
__global__ __launch_bounds__(256) void zero_kernel(float* __restrict__ p, int n) {
    int idx = blockIdx.x * blockDim.x + threadIdx.x;
    if (idx < n) p[idx] = 0.0f;
}

__global__ __launch_bounds__(256) void pool_acc_kernel(
    const float* __restrict__ h, const int* __restrict__ batch,
    float* __restrict__ sums, float* __restrict__ cnt, int total)
{
    int idx = blockIdx.x * blockDim.x + threadIdx.x;
    if (idx >= total) return;                  // total = N * (HID/4)
    int i = idx >> 5;                          // node
    int c = idx & 31;                          // float4 chunk
    int b = batch[i];
    const float4 v = *(const float4*)(h + (size_t)i * HID + 4 * c);
    float* s = sums + (size_t)b * HID + 4 * c;
    unsafeAtomicAdd(s + 0, v.x);
    unsafeAtomicAdd(s + 1, v.y);
    unsafeAtomicAdd(s + 2, v.z);
    unsafeAtomicAdd(s + 3, v.w);
    if (c == 0) unsafeAtomicAdd(cnt + b, 1.0f);
}

__global__ __launch_bounds__(256) void pool_final_kernel(
    const float* __restrict__ sums, const float* __restrict__ cnt,
    float* __restrict__ pooled)
{
    int idx = blockIdx.x * blockDim.x + threadIdx.x;
    if (idx >= GG * HID) return;
    int g = idx / HID;
    pooled[idx] = sums[idx] / fmaxf(cnt[g], 1.0f);
}

// ---- head: fc1 -> LayerNorm -> GELU -> +pooled -> fc2 ----
__global__ __launch_bounds__(128) void head_kernel(
    const float* __restrict__ pooled,
    const float* __restrict__ fc1w, const float* __restrict__ fc1b,
    const float* __restrict__ lng, const float* __restrict__ lnb,
    const float* __restrict__ fc2w, const float* __restrict__ fc2b,
    float* __restrict__ out)
{
    __shared__ float p[HID];
    __shared__ float rbuf[HID];
    __shared__ float red[HID];
    const int g = blockIdx.x;
    const int t = threadIdx.x;

    p[t] = pooled[(size_t)g * HID + t];
    __syncthreads();

    float o = fc1b[t];
    #pragma unroll 4
    for (int k = 0; k < HID; ++k) o += p[k] * fc1w[k * HID + t];

    red[t] = o;
    __syncthreads();
    for (int s = 64; s > 0; s >>= 1) {
        if (t < s) red[t] += red[t + s];
        __syncthreads();
    }
    const float mu = red[0] * (1.0f / HID);
    __syncthreads();

    const float dv = o - mu;
    red[t] = dv * dv;
    __syncthreads();
    for (int s = 64; s > 0; s >>= 1) {
        if (t < s) red[t] += red[t + s];
        __syncthreads();
    }
    const float var = red[0] * (1.0f / HID);

    float y = dv * rsqrtf(var + 1e-5f) * lng[t] + lnb[t];
    y = gelu_exact(y) + p[t];
    rbuf[t] = y;
    __syncthreads();

    if (t < LAT) {
        float o2 = fc2b[t];
        #pragma unroll 4
        for (int k = 0; k < HID; ++k) o2 += rbuf[k] * fc2w[k * LAT + t];
        out[(size_t)g * LAT + t] = o2;
    }
}

extern "C" void kernel_launch(void* const* d_in, const int* in_sizes, int n_in,
                              void* d_out, int out_size, void* d_ws, size_t ws_size,
                              hipStream_t stream) {
    // --- inputs (setup_inputs dict order) ---
    const float* x       = (const float*)d_in[0];
    const int*   ei      = (const int*)  d_in[1];
    const int*   batch   = (const int*)  d_in[2];
    const float* in_g    = (const float*)d_in[3];
    const float* in_b    = (const float*)d_in[4];
    const float* in_m    = (const float*)d_in[5];
    const float* in_v    = (const float*)d_in[6];
    const float* eps0    = (const float*)d_in[7];
    const float* c0_w1   = (const float*)d_in[8];
    const float* c0_b1   = (const float*)d_in[9];
    const float* c0_g    = (const float*)d_in[10];
    const float* c0_bb   = (const float*)d_in[11];
    const float* c0_m    = (const float*)d_in[12];
    const float* c0_v    = (const float*)d_in[13];
    const float* c0_w2   = (const float*)d_in[14];
    const float* c0_b2   = (const float*)d_in[15];
    const float* bn0_g   = (const float*)d_in[16];
    const float* bn0_b   = (const float*)d_in[17];
    const float* bn0_m   = (const float*)d_in[18];
    const float* bn0_v   = (const float*)d_in[19];
    const float* epsR    = (const float*)d_in[20];
    const float* cR_w1   = (const float*)d_in[21];
    const float* cR_b1   = (const float*)d_in[22];
    const float* cR_g    = (const float*)d_in[23];
    const float* cR_bb   = (const float*)d_in[24];
    const float* cR_m    = (const float*)d_in[25];
    const float* cR_v    = (const float*)d_in[26];
    const float* cR_w2   = (const float*)d_in[27];
    const float* cR_b2   = (const float*)d_in[28];
    const float* bnR_g   = (const float*)d_in[29];
    const float* bnR_b   = (const float*)d_in[30];
    const float* bnR_m   = (const float*)d_in[31];
    const float* bnR_v   = (const float*)d_in[32];
    const float* fc1_w   = (const float*)d_in[33];
    const float* fc1_b   = (const float*)d_in[34];
    const float* ln_g    = (const float*)d_in[35];
    const float* ln_b    = (const float*)d_in[36];
    const float* fc2_w   = (const float*)d_in[37];
    const float* fc2_b   = (const float*)d_in[38];

    const int* src = ei;
    const int* dst = ei + EE;

    // --- workspace layout (floats) ---
    float* buf0 = (float*)d_ws;                       // N x 128 (layer output)
    float* buf1 = buf0 + (size_t)NN * HID;            // N x 128 (agg)
    float* buf2 = buf1 + (size_t)NN * HID;            // N x 128 (mid)
    float* sums = buf2 + (size_t)NN * HID;            // G x 128
    float* cnt  = sums + (size_t)GG * HID;            // G

    const int TB = 256;
    const int nTiles = NN / 16;                       // 6250

    // 1) input BN -> buf0 [N][64]; agg init buf1 = (1+eps0)*buf0
    bn_in_kernel<<<(NN * INDIM + TB - 1) / TB, TB, 0, stream>>>(
        x, in_g, in_b, in_m, in_v, eps0, buf0, buf1);

    // 2) layer 0 (64 -> 128); second GEMM pre-initializes next layer's agg
    {
        int total = EE * (INDIM / 4);
        edge_agg_kernel<<<(total + TB - 1) / TB, TB, 0, stream>>>(
            buf0, src, dst, buf1, INDIM, INDIM / 4, total);
        gemm_bn_gelu_kernel<INDIM><<<nTiles, TB, 0, stream>>>(
            buf1, c0_w1, c0_b1, c0_g, c0_bb, c0_m, c0_v, buf2,
            buf1, eps0, 0, 0);
        gemm_bn_gelu_kernel<HID><<<nTiles, TB, 0, stream>>>(
            buf2, c0_w2, c0_b2, bn0_g, bn0_b, bn0_m, bn0_v, buf0,
            buf1, epsR, 0, 1);
    }

    // 3) layers 1..2 (128 -> 128)
    for (int l = 0; l < 2; ++l) {
        const size_t wOff = (size_t)l * HID * HID;
        const size_t vOff = (size_t)l * HID;
        int total = EE * (HID / 4);
        edge_agg_kernel<<<(total + TB - 1) / TB, TB, 0, stream>>>(
            buf0, src, dst, buf1, HID, HID / 4, total);
        gemm_bn_gelu_kernel<HID><<<nTiles, TB, 0, stream>>>(
            buf1, cR_w1 + wOff, cR_b1 + vOff, cR_g + vOff, cR_bb + vOff,
            cR_m + vOff, cR_v + vOff, buf2,
            buf1, epsR, 0, 0);
        // last layer: no next-agg init needed
        const int writeAgg = (l == 0) ? 1 : 0;
        gemm_bn_gelu_kernel<HID><<<nTiles, TB, 0, stream>>>(
            buf2, cR_w2 + wOff, cR_b2 + vOff, bnR_g + vOff, bnR_b + vOff,
            bnR_m + vOff, bnR_v + vOff, buf0,
            buf1, epsR, 1, writeAgg);
    }

    // 4) global mean pool
    zero_kernel<<<(GG * HID + GG + TB - 1) / TB, TB, 0, stream>>>(sums, GG * HID + GG);
    {
        int total = NN * (HID / 4);
        pool_acc_kernel<<<(total + TB - 1) / TB, TB, 0, stream>>>(
            buf0, batch, sums, cnt, total);
    }
    pool_final_kernel<<<(GG * HID + TB - 1) / TB, TB, 0, stream>>>(sums, cnt, buf1);

    // 5) head -> d_out [G][64]
    head_kernel<<<GG, HID, 0, stream>>>(
        buf1, fc1_w, fc1_b, ln_g, ln_b, fc2_w, fc2_b, (float*)d_out);
}